// MSFA_Block_18425409700333
// MI455X (gfx1250) — compile-verified
//
#include <hip/hip_runtime.h>

#define BATCH 4
#define CCH   64
#define HH    192
#define WW    192
#define HWPIX (HH * WW)      // 36864
#define OFFC  18             // 2*3*3 offset channels
#define HID   4              // C/16 squeeze channels
#define BNEPS 1e-5f

typedef float v2f __attribute__((ext_vector_type(2)));
typedef float v8f __attribute__((ext_vector_type(8)));

// D = A(16x4, f32) * B(4x16, f32) + C(16x16, f32)
__device__ __forceinline__ v8f wmma_f32(v2f a, v2f b, v8f c) {
    return __builtin_amdgcn_wmma_f32_16x16x4_f32(
        /*neg_a=*/false, a, /*neg_b=*/false, b,
        /*c_mod=*/(short)0, c, /*reuse_a=*/false, /*reuse_b=*/false);
}

// ---------------------------------------------------------------------------
// K0: zero the pooled accumulator (ws is poisoned by harness)
// ---------------------------------------------------------------------------
__global__ void zero_kernel(float* __restrict__ p, int n) {
    int i = blockIdx.x * blockDim.x + threadIdx.x;
    if (i < n) p[i] = 0.0f;
}

// ---------------------------------------------------------------------------
// K1: offset-prediction conv (3x3, C=64 -> 18) as WMMA GEMM.
//     K ordered tap-major (K = tap*64 + c). Rolled ti/tj loops (no division),
//     branch-free masked loads, pointer-bump inner loop with capped unroll
//     to keep the register allocator away from scratch spills.
//     block = 128 threads = 4 waves; each wave: one 16-pixel tile, 2 M-tiles.
// ---------------------------------------------------------------------------
__global__ void __launch_bounds__(128)
offset_conv_kernel(const float* __restrict__ x,
                   const float* __restrict__ off_w,
                   const float* __restrict__ off_b,
                   float* __restrict__ off_out) {
    const int wave = threadIdx.x >> 5;
    const int lane = threadIdx.x & 31;
    const int half = lane >> 4;      // 0 -> K+0/1 ; 1 -> K+2/3
    const int l16  = lane & 15;      // M row / N column within tile

    const int tile = blockIdx.x * 64 + wave * 16;  // linear pixel over B*HW
    const int b  = tile / HWPIX;
    const int p0 = tile % HWPIX;                   // 16 | 192: never crosses rows
    const int y  = p0 / WW;
    const int x0 = p0 % WW;

    const float* xb = x + (long)b * CCH * HWPIX;

    // M-tile 1 covers channels 16..31; only 16,17 are real.
    const int   m1row  = min(16 + l16, OFFC - 1);
    const float m1mask = (16 + l16 < OFFC) ? 1.0f : 0.0f;

    v8f acc0 = {0,0,0,0,0,0,0,0};
    v8f acc1 = {0,0,0,0,0,0,0,0};

    int tap = 0;
#pragma unroll 1
    for (int ti = 0; ti < 3; ++ti) {
#pragma unroll 1
        for (int tj = 0; tj < 3; ++tj, ++tap) {
            const int yy = y + ti - 1;
            const int xx = x0 + l16 + tj - 1;
            const float msk =
                (yy >= 0 && yy < HH && xx >= 0 && xx < WW) ? 1.0f : 0.0f;
            const int yyc = min(max(yy, 0), HH - 1);
            const int xxc = min(max(xx, 0), WW - 1);

            // lane starts at channel c = half*2; step 4 channels per iter
            const float* xcol  = xb + yyc * WW + xxc + (long)(half * 2) * HWPIX;
            const float* wrow0 = off_w + l16   * 576 + (half * 2) * 9 + tap;
            const float* wrow1 = off_w + m1row * 576 + (half * 2) * 9 + tap;

#pragma unroll 2
            for (int c0 = 0; c0 < CCH; c0 += 4) {
                v2f bv, av0, av1;
                bv.x  = msk * xcol[0];
                bv.y  = msk * xcol[HWPIX];
                av0.x = wrow0[0];
                av0.y = wrow0[9];
                av1.x = m1mask * wrow1[0];
                av1.y = m1mask * wrow1[9];
                acc0 = wmma_f32(av0, bv, acc0);
                acc1 = wmma_f32(av1, bv, acc1);
                xcol  += 4 * HWPIX;
                wrow0 += 36;
                wrow1 += 36;
            }
        }
    }

    float* ob = off_out + (long)b * OFFC * HWPIX;
    const int pix = p0 + l16;
    for (int r = 0; r < 8; ++r) {
        int ch0 = r + half * 8;              // D layout: VGPR r = rows r / r+8
        if (ch0 < OFFC) ob[ch0 * HWPIX + pix] = acc0[r] + off_b[ch0];
        int ch1 = 16 + r + half * 8;
        if (ch1 < OFFC) ob[ch1 * HWPIX + pix] = acc1[r] + off_b[ch1];
    }
}

// ---------------------------------------------------------------------------
// K2: deformable conv (9 taps, 64->64) + bias + BN, with pooled-sum atomics.
//     block = 128 threads = 4 waves, one 16-pixel tile per block.
//     Per tap: stage bilinear samples [64ch x 16pix] in LDS (WMMA-B layout),
//     then each wave (one 16-channel M-tile) runs 16 wmma_f32 ops.
// ---------------------------------------------------------------------------
__global__ void __launch_bounds__(128)
deform_conv_bn_kernel(const float* __restrict__ x,
                      const float* __restrict__ off,
                      const float* __restrict__ dw,
                      const float* __restrict__ db,
                      const float* __restrict__ bn_g,
                      const float* __restrict__ bn_b,
                      const float* __restrict__ bn_m,
                      const float* __restrict__ bn_v,
                      float* __restrict__ h_out,
                      float* __restrict__ pooled) {
    __shared__ float lds_samp[CCH * 16];   // samp[c][n]
    __shared__ int   ls_y0[16], ls_x0[16];
    __shared__ float ls_wy[16], ls_wx[16];

    const int tid  = threadIdx.x;
    const int wave = tid >> 5;
    const int lane = tid & 31;
    const int half = lane >> 4;
    const int l16  = lane & 15;
    const int mbase = wave * 16;

    const int plin = blockIdx.x * 16;
    const int b   = plin / HWPIX;
    const int p0  = plin % HWPIX;
    const int y   = p0 / WW;
    const int xx0 = p0 % WW;

    const float* xb   = x   + (long)b * CCH  * HWPIX;
    const float* offb = off + (long)b * OFFC * HWPIX;

    v8f acc = {0,0,0,0,0,0,0,0};

#pragma unroll 1
    for (int tap = 0; tap < 9; ++tap) {
        if (tid < 16) {
            const int i = tap / 3, j = tap % 3;   // once per tap: cheap
            float dy = offb[(2 * tap)     * HWPIX + p0 + tid];
            float dx = offb[(2 * tap + 1) * HWPIX + p0 + tid];
            float py = (float)(y + i - 1) + dy;
            float px = (float)(xx0 + tid + j - 1) + dx;
            float fy = floorf(py), fx = floorf(px);
            ls_y0[tid] = (int)fy;  ls_x0[tid] = (int)fx;
            ls_wy[tid] = py - fy;  ls_wx[tid] = px - fx;
        }
        __syncthreads();

        { // gather: thread -> (n = tid&15, channels cb..cb+7); branch-free masks
            const int n  = tid & 15;
            const int cb = (tid >> 4) * 8;
            const int y0 = ls_y0[n], x0i = ls_x0[n];
            const float wy = ls_wy[n], wx = ls_wx[n];
            const int y0c = min(max(y0,     0), HH - 1);
            const int y1c = min(max(y0 + 1, 0), HH - 1);
            const int x0c = min(max(x0i,     0), WW - 1);
            const int x1c = min(max(x0i + 1, 0), WW - 1);
            const float ym0 = (y0 >= 0     && y0 < HH)     ? 1.0f : 0.0f;
            const float ym1 = (y0 + 1 >= 0 && y0 + 1 < HH) ? 1.0f : 0.0f;
            const float xm0 = (x0i >= 0     && x0i < WW)     ? 1.0f : 0.0f;
            const float xm1 = (x0i + 1 >= 0 && x0i + 1 < WW) ? 1.0f : 0.0f;
            const float w00 = (1.f - wy) * (1.f - wx) * ym0 * xm0;
            const float w01 = (1.f - wy) * wx         * ym0 * xm1;
            const float w10 = wy * (1.f - wx)         * ym1 * xm0;
            const float w11 = wy * wx                 * ym1 * xm1;
            const int i00 = y0c * WW + x0c, i01 = y0c * WW + x1c;
            const int i10 = y1c * WW + x0c, i11 = y1c * WW + x1c;
            const float* xp = xb + (long)cb * HWPIX;
            float* lp = &lds_samp[cb * 16 + n];
#pragma unroll
            for (int q = 0; q < 8; ++q) {
                lp[q * 16] = w00 * xp[i00] + w01 * xp[i01]
                           + w10 * xp[i10] + w11 * xp[i11];
                xp += HWPIX;
            }
        }
        __syncthreads();

        // GEMM: K=64 in steps of 4; pointer-bump addressing
        {
            const float* ap = dw + ((long)(mbase + l16) * CCH + half * 2) * 9 + tap;
            const float* bp = &lds_samp[(half * 2) * 16 + l16];
#pragma unroll 4
            for (int k0 = 0; k0 < CCH; k0 += 4) {
                v2f av, bv;
                av.x = ap[0];
                av.y = ap[9];
                bv.x = bp[0];
                bv.y = bp[16];
                acc = wmma_f32(av, bv, acc);
                ap += 36;
                bp += 64;
            }
        }
        __syncthreads();   // protect lds_samp before next tap's gather
    }

    // epilogue: bias + BN, store h, accumulate global-avg-pool sums
    float* hb = h_out + (long)b * CCH * HWPIX;
    for (int r = 0; r < 8; ++r) {
        const int m = mbase + r + half * 8;
        const float inv = rsqrtf(bn_v[m] + BNEPS);
        float val = (acc[r] + db[m] - bn_m[m]) * (bn_g[m] * inv) + bn_b[m];
        hb[(long)m * HWPIX + p0 + l16] = val;
        // reduce over the 16-lane half (xor 1,2,4,8 stays within the half)
        float s = val;
        s += __shfl_xor(s, 1, 32);
        s += __shfl_xor(s, 2, 32);
        s += __shfl_xor(s, 4, 32);
        s += __shfl_xor(s, 8, 32);
        if (l16 == 0) atomicAdd(&pooled[b * CCH + m], s);
    }
}

// ---------------------------------------------------------------------------
// K3: channel attention MLP (64 -> 4 -> 64), sigmoid. 4 blocks x 64 threads.
// ---------------------------------------------------------------------------
__global__ void ca_kernel(const float* __restrict__ pooled,
                          const float* __restrict__ w1, const float* __restrict__ b1,
                          const float* __restrict__ w2, const float* __restrict__ b2,
                          float* __restrict__ cw) {
    __shared__ float mean[CCH];
    __shared__ float hid[HID];
    const int b = blockIdx.x, t = threadIdx.x;
    mean[t] = pooled[b * CCH + t] * (1.0f / (float)HWPIX);
    __syncthreads();
    if (t < HID) {
        float s = b1[t];
        for (int c = 0; c < CCH; ++c) s += w1[t * CCH + c] * mean[c];
        hid[t] = fmaxf(s, 0.0f);
    }
    __syncthreads();
    float s = b2[t];
    for (int h = 0; h < HID; ++h) s += w2[t * HID + h] * hid[h];
    cw[b * CCH + t] = 1.0f / (1.0f + expf(-s));
}

// ---------------------------------------------------------------------------
// K4: spatial attention (7x7 conv over h*cw -> sigmoid) fused with both
//     elementwise multiplies. Channel scale folded into per-block LDS weights.
//     c-outer loops give contiguous 7-wide row loads; interior fast path is
//     branch-free. grid = (HW/256, B), block = 256.
// ---------------------------------------------------------------------------
__global__ void sa_out_kernel(const float* __restrict__ h,
                              const float* __restrict__ cw,
                              const float* __restrict__ sa_w,
                              const float* __restrict__ sa_b,
                              float* __restrict__ out) {
    __shared__ float wcw[CCH * 49];
    const int b = blockIdx.y;
    const int t = threadIdx.x;
    for (int i = t; i < CCH * 49; i += blockDim.x) {
        const int c = i / 49;
        wcw[i] = sa_w[i] * cw[b * CCH + c];
    }
    __syncthreads();

    const int p = blockIdx.x * blockDim.x + t;   // HW == 144*256 exactly
    const int y = p / WW, xp = p % WW;
    const float* hb = h + (long)b * CCH * HWPIX;

    float sum = sa_b[0];
    if (y >= 3 && y < HH - 3 && xp >= 3 && xp < WW - 3) {
        // interior: no bounds checks, contiguous row access
        const float* hp = hb + (y - 3) * WW + (xp - 3);
        const float* wp = wcw;
#pragma unroll 1
        for (int c = 0; c < CCH; ++c) {
#pragma unroll
            for (int dy = 0; dy < 7; ++dy) {
                const float* row = hp + dy * WW;
#pragma unroll
                for (int dx = 0; dx < 7; ++dx)
                    sum += row[dx] * wp[dy * 7 + dx];
            }
            hp += HWPIX;
            wp += 49;
        }
    } else {
#pragma unroll 1
        for (int c = 0; c < CCH; ++c) {
            const float* hp = hb + (long)c * HWPIX;
            const float* wp = wcw + c * 49;
            for (int dy = 0; dy < 7; ++dy) {
                const int yy = y + dy - 3;
                if (yy < 0 || yy >= HH) continue;
                const float* row = hp + yy * WW;
                for (int dx = 0; dx < 7; ++dx) {
                    const int xx = xp + dx - 3;
                    if (xx < 0 || xx >= WW) continue;
                    sum += row[xx] * wp[dy * 7 + dx];
                }
            }
        }
    }
    const float sw = 1.0f / (1.0f + expf(-sum));
    float* ob = out + (long)b * CCH * HWPIX;
    for (int c = 0; c < CCH; ++c)
        ob[(long)c * HWPIX + p] = hb[(long)c * HWPIX + p] * cw[b * CCH + c] * sw;
}

// ---------------------------------------------------------------------------
extern "C" void kernel_launch(void* const* d_in, const int* in_sizes, int n_in,
                              void* d_out, int out_size, void* d_ws, size_t ws_size,
                              hipStream_t stream) {
    const float* x      = (const float*)d_in[0];
    const float* off_w  = (const float*)d_in[1];
    const float* off_b  = (const float*)d_in[2];
    const float* dw     = (const float*)d_in[3];
    const float* db     = (const float*)d_in[4];
    const float* bn_g   = (const float*)d_in[5];
    const float* bn_b   = (const float*)d_in[6];
    const float* bn_m   = (const float*)d_in[7];
    const float* bn_v   = (const float*)d_in[8];
    const float* ca_w1  = (const float*)d_in[9];
    const float* ca_b1  = (const float*)d_in[10];
    const float* ca_w2  = (const float*)d_in[11];
    const float* ca_b2  = (const float*)d_in[12];
    const float* sa_w   = (const float*)d_in[13];
    const float* sa_b   = (const float*)d_in[14];
    float* out = (float*)d_out;

    // workspace carve-up (all fp32): offsets | h | pooled | cw  (~48.4 MB)
    float* off_ws = (float*)d_ws;
    float* h_ws   = off_ws + (size_t)BATCH * OFFC * HWPIX;
    float* pooled = h_ws   + (size_t)BATCH * CCH  * HWPIX;
    float* cw     = pooled + BATCH * CCH;

    zero_kernel<<<1, 256, 0, stream>>>(pooled, BATCH * CCH);

    offset_conv_kernel<<<(BATCH * HWPIX) / 64, 128, 0, stream>>>(x, off_w, off_b, off_ws);

    deform_conv_bn_kernel<<<(BATCH * HWPIX) / 16, 128, 0, stream>>>(
        x, off_ws, dw, db, bn_g, bn_b, bn_m, bn_v, h_ws, pooled);

    ca_kernel<<<BATCH, CCH, 0, stream>>>(pooled, ca_w1, ca_b1, ca_w2, ca_b2, cw);

    sa_out_kernel<<<dim3(HWPIX / 256, BATCH), 256, 0, stream>>>(h_ws, cw, sa_w, sa_b, out);
}